// Detection_15461882266079
// MI455X (gfx1250) — compile-verified
//
#include <hip/hip_runtime.h>
#include <hip/hip_bf16.h>

// Greedy class-aware weighted boxes fusion (WBF) + top-1000, single workgroup.
// All cluster state lives in LDS (~144KB -- only possible on CDNA5's 320KB WGP LDS).
// Sequential scan over 4096 dets; inner IoU test parallel over 512 lanes (16 wave32).
// Next detection row is double-buffered into LDS via CDNA5 async global->LDS DMA.

#define NDET   4096
#define NT     512
#define NPOST  1000
#define IOU_T  0.55f
#define SENT   0x7fffffff

#if defined(__HIP_DEVICE_COMPILE__) && __has_builtin(__builtin_amdgcn_global_load_async_to_lds_b32)
#define HAS_ASYNC_LDS 1
#else
#define HAS_ASYNC_LDS 0
#endif

typedef __attribute__((address_space(1))) int* gint_ptr;
typedef __attribute__((address_space(3))) int* lint_ptr;

__device__ __forceinline__ void det_prefetch_word(const float* g, float* l) {
#if HAS_ASYNC_LDS
  // global_load_async_to_lds_b32: per-lane DMA of 4 bytes, tracked by ASYNCcnt.
  __builtin_amdgcn_global_load_async_to_lds_b32(
      (gint_ptr)(unsigned long long)(g),
      (lint_ptr)(unsigned)(unsigned long long)(l),
      /*imm offset*/ 0, /*cpol*/ 0);
#else
  *l = *g;
#endif
}

__device__ __forceinline__ void det_prefetch_wait() {
#if HAS_ASYNC_LDS
  asm volatile("s_wait_asynccnt 0" ::: "memory");
#endif
}

__global__ __launch_bounds__(NT, 1) void wbf_topk_kernel(
    const float* __restrict__ boxes,     // [16,256,6]
    const float* __restrict__ offsets,   // [16,2]
    float* __restrict__ out)             // [1000,6]
{
  __shared__ float4 s_w[NDET];     // wsum during scan; mbox after scan   (64KB)
  __shared__ float  s_cls[NDET];   // cluster class, -1 = empty           (16KB)
  __shared__ float  s_ssum[NDET];  // score sum                           (16KB)
  __shared__ float  s_cnt[NDET];   // member count                        (16KB)
  __shared__ float  s_key[NDET];   // sort keys                           (16KB)
  __shared__ int    s_sidx[NDET];  // sort payload (cluster index)        (16KB)
  __shared__ float  s_det[2][8];   // double-buffered raw detection row
  __shared__ float  s_off[32];     // offsets copy
  __shared__ int    s_min;         // min matching cluster index this step
  __shared__ int    s_num;         // number of clusters

  const int tid = threadIdx.x;

  // ---- init ----
  for (int j = tid; j < NDET; j += NT) {
    s_w[j]    = make_float4(0.f, 0.f, 0.f, 0.f);
    s_cls[j]  = -1.0f;
    s_ssum[j] = 0.0f;
    s_cnt[j]  = 0.0f;
  }
  if (tid < 32) s_off[tid] = offsets[tid];
  if (tid < 6)  s_det[0][tid] = boxes[tid];    // raw det row 0
  if (tid == 0) { s_min = SENT; s_num = 0; }
  __syncthreads();

  // ---- sequential greedy WBF scan ----
  for (int i = 0; i < NDET; ++i) {
    const int cb = i & 1, nb = cb ^ 1;
    const int p  = i >> 8;                      // patch index (K=256)
    const float ox = s_off[2*p + 0];
    const float oy = s_off[2*p + 1];
    // current detection (uniform across wave; SIZE_FACTOR == 1.0)
    const float bx1 = s_det[cb][0] + ox;
    const float by1 = s_det[cb][1] + oy;
    const float bx2 = s_det[cb][2] + ox;
    const float by2 = s_det[cb][3] + oy;
    const float sc  = s_det[cb][4];
    const float lb  = s_det[cb][5];
    const float areaA = (bx2 - bx1) * (by2 - by1);

    // kick off async DMA of next raw det row into the other buffer
    const int pi = (i + 1 < NDET) ? (i + 1) : (NDET - 1);
    if (tid < 6) det_prefetch_word(boxes + pi * 6 + tid, &s_det[nb][tid]);

    // parallel IoU test against existing clusters; find first match
    const int num = s_num;
    int local = SENT;
    for (int j = tid; j < num; j += NT) {
      if (s_cls[j] == lb) {                     // cls set => cnt>0
        const float4 w  = s_w[j];
        const float inv = 1.0f / s_ssum[j];
        const float mx1 = w.x * inv, my1 = w.y * inv;
        const float mx2 = w.z * inv, my2 = w.w * inv;
        const float xx1 = fmaxf(mx1, bx1), yy1 = fmaxf(my1, by1);
        const float xx2 = fminf(mx2, bx2), yy2 = fminf(my2, by2);
        const float inter = fmaxf(xx2 - xx1, 0.f) * fmaxf(yy2 - yy1, 0.f);
        const float areaB = (mx2 - mx1) * (my2 - my1);
        const float uni   = areaA + areaB - inter;
        if (uni > 0.f && inter > IOU_T * uni) { local = j; break; }  // j increasing
      }
    }
    if (local != SENT) atomicMin(&s_min, local);
    __syncthreads();

    // single-thread cluster update (matches reference update order)
    if (tid == 0) {
      const int m = s_min;
      int idx;
      if (m < NDET) idx = m;
      else { idx = s_num; s_num = idx + 1; }
      float4 w = s_w[idx];
      w.x += sc * bx1; w.y += sc * by1; w.z += sc * bx2; w.w += sc * by2;
      s_w[idx]     = w;
      s_ssum[idx] += sc;
      s_cnt[idx]  += 1.0f;
      s_cls[idx]   = lb;
      s_min        = SENT;
    }
    det_prefetch_wait();    // ASYNCcnt==0: prefetched row is in LDS
    __syncthreads();        // publish cluster update + prefetched det
  }

  // ---- finalize: mbox = wsum/ssum (exact), sort keys ----
  for (int j = tid; j < NDET; j += NT) {
    const float c = s_cnt[j];
    float key = -1.0f;
    if (c > 0.f) {
      const float ss = s_ssum[j];
      float4 w = s_w[j];
      w.x /= ss; w.y /= ss; w.z /= ss; w.w /= ss;
      s_w[j] = w;
      key = ss / c;         // == ssum / max(cnt,1) for valid clusters
    }
    s_key[j]  = key;
    s_sidx[j] = j;
  }
  __syncthreads();

  // ---- bitonic sort, descending by key, tie -> smaller index (stable top_k) ----
  for (int k = 2; k <= NDET; k <<= 1) {
    for (int jj = k >> 1; jj > 0; jj >>= 1) {
      for (int i = tid; i < NDET; i += NT) {
        const int l = i ^ jj;
        if (l > i) {
          const float ki = s_key[i], kl = s_key[l];
          const int   ii = s_sidx[i], il = s_sidx[l];
          const bool prec_l_i = (kl > ki) || (kl == ki && il < ii);
          const bool prec_i_l = (ki > kl) || (ki == kl && ii < il);
          const bool doswap = ((i & k) == 0) ? prec_l_i : prec_i_l;
          if (doswap) {
            s_key[i] = kl; s_key[l] = ki;
            s_sidx[i] = il; s_sidx[l] = ii;
          }
        }
      }
      __syncthreads();
    }
  }

  // ---- emit top-1000 rows: [x1,y1,x2,y2,score,label] or zeros ----
  for (int r = tid; r < NPOST; r += NT) {
    const float key = s_key[r];
    float o0 = 0.f, o1 = 0.f, o2 = 0.f, o3 = 0.f, o4 = 0.f, o5 = 0.f;
    if (key >= 0.f) {
      const int j = s_sidx[r];
      const float4 w = s_w[j];
      o0 = w.x; o1 = w.y; o2 = w.z; o3 = w.w;
      o4 = key; o5 = s_cls[j];
    }
    float* o = out + r * 6;
    o[0] = o0; o[1] = o1; o[2] = o2; o[3] = o3; o[4] = o4; o[5] = o5;
  }
}

extern "C" void kernel_launch(void* const* d_in, const int* in_sizes, int n_in,
                              void* d_out, int out_size, void* d_ws, size_t ws_size,
                              hipStream_t stream) {
  (void)in_sizes; (void)n_in; (void)d_ws; (void)ws_size; (void)out_size;
  const float* boxes   = (const float*)d_in[0];   // [16,256,6] f32
  const float* offsets = (const float*)d_in[1];   // [16,2]     f32
  float*       out     = (float*)d_out;           // [1000,6]   f32
  wbf_topk_kernel<<<1, NT, 0, stream>>>(boxes, offsets, out);
}